// PCPiecewiseRigid_62612033241842
// MI455X (gfx1250) — compile-verified
//
#include <hip/hip_runtime.h>
#include <hip/hip_bf16.h>
#include <math.h>

// Problem constants (match reference)
constexpr int BB   = 16;
constexpr int NN   = 2048;
constexpr int HID_ = 32;
constexpr int NSEG_ = 3;
constexpr int NTILES = NN / 16;          // 128 tiles of 16 per dimension
constexpr float PI_F = 3.14159265358979323846f;

typedef __attribute__((ext_vector_type(2))) float v2f;
typedef __attribute__((ext_vector_type(8))) float v8f;

// ---------------------------------------------------------------------------
// Kernel 1: h = relu(pos_flat @ W1^T + b1)   -> ws.h  (16 x 32)
// pos_flat(b) is exactly the first 3*N contiguous floats of x[b] (x is B,6,N)
// ---------------------------------------------------------------------------
__global__ void k_hidden(const float* __restrict__ x, const float* __restrict__ W1,
                         const float* __restrict__ b1, float* __restrict__ h) {
    int b = blockIdx.x / HID_;
    int j = blockIdx.x % HID_;
    const float* xb = x + (size_t)b * 6 * NN;
    const float* w  = W1 + (size_t)j * 3 * NN;
    float s = 0.f;
    for (int k = threadIdx.x; k < 3 * NN; k += blockDim.x)
        s += xb[k] * w[k];
    __shared__ float red[256];
    red[threadIdx.x] = s;
    __syncthreads();
    for (int off = 128; off > 0; off >>= 1) {
        if ((int)threadIdx.x < off) red[threadIdx.x] += red[threadIdx.x + off];
        __syncthreads();
    }
    if (threadIdx.x == 0) h[b * HID_ + j] = fmaxf(red[0] + b1[j], 0.f);
}

// ---------------------------------------------------------------------------
// Kernel 2: z = tanh(h @ W2^T + b2); build per-(b,seg) R (3x3) and T (3)
// ---------------------------------------------------------------------------
__device__ inline void mat3mul(float* __restrict__ d, const float* __restrict__ a,
                               const float* __restrict__ b) {
    for (int m = 0; m < 3; ++m)
        for (int s = 0; s < 3; ++s)
            d[m * 3 + s] = a[m * 3 + 0] * b[0 * 3 + s]
                         + a[m * 3 + 1] * b[1 * 3 + s]
                         + a[m * 3 + 2] * b[2 * 3 + s];
}

__global__ void k_rigid(const float* __restrict__ h, const float* __restrict__ W2,
                        const float* __restrict__ b2, const float* __restrict__ center,
                        const float* __restrict__ sz, float* __restrict__ Rm,
                        float* __restrict__ Tv) {
    __shared__ float z[BB][6 * NSEG_];
    for (int idx = threadIdx.x; idx < BB * 6 * NSEG_; idx += blockDim.x) {
        int b = idx / (6 * NSEG_), e = idx % (6 * NSEG_);
        float s = b2[e];
        for (int k = 0; k < HID_; ++k) s += h[b * HID_ + k] * W2[e * HID_ + k];
        z[b][e] = tanhf(s);
    }
    __syncthreads();
    for (int idx = threadIdx.x; idx < BB * NSEG_; idx += blockDim.x) {
        int b = idx / NSEG_, sg = idx % NSEG_;
        float t0 = z[b][sg * 6 + 3], t1 = z[b][sg * 6 + 4], t2 = z[b][sg * 6 + 5];
        float a0 = PI_F * t0, a1 = PI_F * t1, a2 = PI_F * t2;
        float c0 = cosf(a0), s0 = sinf(a0);
        float c1 = cosf(a1), s1 = sinf(a1);
        float c2 = cosf(a2), s2 = sinf(a2);
        float X[9]  = {c0, -s0, 0.f,  s0, c0, 0.f,  0.f, 0.f, 1.f};
        float Y[9]  = {c1, 0.f, s1,   0.f, 1.f, 0.f, -s1, 0.f, c1};
        float Zm[9] = {1.f, 0.f, 0.f, 0.f, c2, -s2,  0.f, s2, c2};
        float M1[9], R[9];
        mat3mul(M1, X, Y);
        mat3mul(R, M1, Zm);
        float tr[3] = {t0, t1, t2};
        float* Rout = Rm + (size_t)(b * NSEG_ + sg) * 9;
        float* Tout = Tv + (size_t)(b * NSEG_ + sg) * 3;
        for (int m = 0; m < 9; ++m) Rout[m] = R[m];
        for (int s = 0; s < 3; ++s) {
            float acc = tr[s] * sz[s];
            for (int m = 0; m < 3; ++m)
                acc += center[m] * (((m == s) ? 1.f : 0.f) - R[m * 3 + s]);
            Tout[s] = acc;
        }
    }
}

// ---------------------------------------------------------------------------
// Kernel 3: pos_t[b,t,n] = sum_k R[b,seg(n)][k][t]*pos[b,k,n] + T[b,seg(n)][t]
//           colors = softmax over channels 3..5; write X_t (B,6,N) to d_out
// ---------------------------------------------------------------------------
__global__ void k_points(const float* __restrict__ x, const float* __restrict__ Rm,
                         const float* __restrict__ Tv, float* __restrict__ out) {
    int gid = blockIdx.x * blockDim.x + threadIdx.x;
    if (gid >= BB * NN) return;
    int b = gid / NN, n = gid % NN;
    int sg = (n < 20) ? 0 : ((n < 40) ? 1 : 2);
    const float* xb = x + (size_t)b * 6 * NN;
    float p0 = xb[n], p1 = xb[NN + n], p2 = xb[2 * NN + n];
    const float* R = Rm + (size_t)(b * NSEG_ + sg) * 9;
    const float* T = Tv + (size_t)(b * NSEG_ + sg) * 3;
    float* ob = out + (size_t)b * 6 * NN;
    for (int t = 0; t < 3; ++t)
        ob[t * NN + n] = R[0 * 3 + t] * p0 + R[1 * 3 + t] * p1 + R[2 * 3 + t] * p2 + T[t];
    float c0 = xb[3 * NN + n], c1 = xb[4 * NN + n], c2 = xb[5 * NN + n];
    float m = fmaxf(c0, fmaxf(c1, c2));
    float e0 = expf(c0 - m), e1 = expf(c1 - m), e2 = expf(c2 - m);
    float inv = 1.f / (e0 + e1 + e2);
    ob[3 * NN + n] = e0 * inv;
    ob[4 * NN + n] = e1 * inv;
    ob[5 * NN + n] = e2 * inv;
}

// ---------------------------------------------------------------------------
// Branchless reference-matching distance: where(d2>0, sqrt(max(d2,1e-12)), 0)
// after clamping d2 at 0. __builtin_amdgcn_sqrtf -> raw v_sqrt_f32 (no Newton
// fixup, no exec-mask branches); inputs clamped to >=1e-12 (normal range).
// ---------------------------------------------------------------------------
__device__ inline float dist_ref(float d2) {
    d2 = fmaxf(d2, 0.f);
    float s = __builtin_amdgcn_sqrtf(fmaxf(d2, 1e-12f));
    return (d2 > 0.f) ? s : 0.f;
}

// ---------------------------------------------------------------------------
// Kernel 4: fused reg reduction. Grid = B*128 blocks (b, i-tile), 8 waves.
// diff^2 is symmetric in (i,j) (diagonal is exactly 0 for both dA and dx),
// so each block only walks j-tiles jt >= ib; off-diagonal tiles weighted x2.
// Per 16x16 tile: two V_WMMA_F32_16X16X4_F32 Gram matrices (pos and A, K=3
// zero-padded to 4), then branchless sqrt epilogue.
// Deterministic: shuffle reduce -> LDS wave reduce -> per-block partial.
// ---------------------------------------------------------------------------
__global__ void k_reg(const float* __restrict__ x, const float* __restrict__ A,
                      float* __restrict__ part) {
    int b    = blockIdx.x / NTILES;
    int ib   = blockIdx.x % NTILES;
    int lane = threadIdx.x & 31;
    int wave = threadIdx.x >> 5;
    int col  = lane & 15;
    bool hi  = lane >= 16;
    const float* px = x + (size_t)b * 6 * NN;   // pos channels 0..2

    // ---- i-tile fragments (fixed per block) -------------------------------
    int i = ib * 16 + col;
    float xi0 = px[i], xi1 = px[NN + i], xi2 = px[2 * NN + i];
    float ai0 = A[i],  ai1 = A[NN + i],  ai2 = A[2 * NN + i];
    // ISA 16x4 f32 A layout: lanes 0-15 carry K={0,1}, lanes 16-31 carry K={2,3}
    v2f fax = hi ? (v2f){xi2, 0.f} : (v2f){xi0, xi1};
    v2f faA = hi ? (v2f){ai2, 0.f} : (v2f){ai0, ai1};
    float nix = xi0 * xi0 + xi1 * xi1 + xi2 * xi2;
    float niA = ai0 * ai0 + ai1 * ai1 + ai2 * ai2;
    // Row norms per accumulator VGPR r: row M = r + 8*(lane/16)
    float sqix[8], sqiA[8];
    int base_m = hi ? 8 : 0;
#pragma unroll
    for (int r = 0; r < 8; ++r) {
        sqix[r] = __shfl(nix, base_m + r, 32);
        sqiA[r] = __shfl(niA, base_m + r, 32);
    }

    float acc = 0.f;
    // trip condition is wave-uniform -> EXEC all-ones at every WMMA
    for (int jt = ib + wave; jt < NTILES; jt += 8) {
        int j = jt * 16 + col;
        float xj0 = px[j], xj1 = px[NN + j], xj2 = px[2 * NN + j];
        float aj0 = A[j],  aj1 = A[NN + j],  aj2 = A[2 * NN + j];
        v2f fbx = hi ? (v2f){xj2, 0.f} : (v2f){xj0, xj1};
        v2f fbA = hi ? (v2f){aj2, 0.f} : (v2f){aj0, aj1};
        float njx = xj0 * xj0 + xj1 * xj1 + xj2 * xj2;
        float njA = aj0 * aj0 + aj1 * aj1 + aj2 * aj2;
        v8f cz = {};
        v8f gx = __builtin_amdgcn_wmma_f32_16x16x4_f32(false, fax, false, fbx,
                                                       (short)0, cz, false, false);
        v8f gA = __builtin_amdgcn_wmma_f32_16x16x4_f32(false, faA, false, fbA,
                                                       (short)0, cz, false, false);
        float ts = 0.f;
#pragma unroll
        for (int r = 0; r < 8; ++r) {
            float dxv = dist_ref(sqix[r] + njx - 2.f * gx[r]);
            float dav = dist_ref(sqiA[r] + njA - 2.f * gA[r]);
            float diff = dav - dxv;
            ts += diff * diff;
        }
        acc += (jt == ib) ? ts : 2.f * ts;   // mirror tile (jt,ib) by symmetry
    }
    // deterministic wave reduce
#pragma unroll
    for (int off = 16; off > 0; off >>= 1) acc += __shfl_xor(acc, off, 32);
    __shared__ float wsum[8];
    if (lane == 0) wsum[wave] = acc;
    __syncthreads();
    if (threadIdx.x == 0) {
        float s = 0.f;
        for (int w = 0; w < 8; ++w) s += wsum[w];
        part[b * NTILES + ib] = s;
    }
}

// ---------------------------------------------------------------------------
// Kernel 5: reg[b] = sqrt(sum of 128 partials), fixed order
// ---------------------------------------------------------------------------
__global__ void k_final(const float* __restrict__ part, float* __restrict__ reg) {
    int b = threadIdx.x;
    if (b < BB) {
        float s = 0.f;
        for (int k = 0; k < NTILES; ++k) s += part[b * NTILES + k];
        reg[b] = sqrtf(s);
    }
}

// ---------------------------------------------------------------------------
extern "C" void kernel_launch(void* const* d_in, const int* in_sizes, int n_in,
                              void* d_out, int out_size, void* d_ws, size_t ws_size,
                              hipStream_t stream) {
    const float* x      = (const float*)d_in[0];   // (16,6,2048)
    const float* W1     = (const float*)d_in[1];   // (32,6144)
    const float* b1     = (const float*)d_in[2];   // (32,)
    const float* W2     = (const float*)d_in[3];   // (18,32)
    const float* b2     = (const float*)d_in[4];   // (18,)
    const float* A      = (const float*)d_in[5];   // (6,2048)
    const float* center = (const float*)d_in[6];   // (3,)
    const float* sz     = (const float*)d_in[7];   // (3,)
    (void)in_sizes; (void)n_in; (void)out_size; (void)ws_size;

    float* out = (float*)d_out;                    // X_t (16*6*2048) then reg (16)
    float* ws  = (float*)d_ws;
    float* h    = ws;                              // 512 floats
    float* Rm   = ws + 512;                        // 432 floats
    float* Tv   = ws + 1024;                       // 144 floats
    float* part = ws + 2048;                       // 2048 floats

    k_hidden<<<BB * HID_, 256, 0, stream>>>(x, W1, b1, h);
    k_rigid <<<1, 128, 0, stream>>>(h, W2, b2, center, sz, Rm, Tv);
    k_points<<<(BB * NN + 255) / 256, 256, 0, stream>>>(x, Rm, Tv, out);
    k_reg   <<<BB * NTILES, 256, 0, stream>>>(x, A, part);
    k_final <<<1, 32, 0, stream>>>(part, out + (size_t)BB * 6 * NN);
}